// FWHT_83476984365427
// MI455X (gfx1250) — compile-verified
//
#include <hip/hip_runtime.h>
#include <stdint.h>

// FWHT-4096 via H_4096 = H_16 (x) H_16 (x) H_16 on the CDNA5 matrix pipe.
// Three batched 16x16x16 GEMM stages per row, four chained
// V_WMMA_F32_16X16X4_F32 per tile (full f32 precision).
//
// One wave32 per row, wave-private LDS slice (no barriers; per-wave LDS is
// in-order). Inbound row copy uses GLOBAL_LOAD_ASYNC_TO_LDS_B128 (ASYNCcnt).
// Stage 3 output is lane-contiguous, so it stores directly to global with the
// 1/sqrt(4096) scale folded into a pre-scaled B fragment (exact: 1/64).

typedef float v2f __attribute__((ext_vector_type(2)));
typedef float v8f __attribute__((ext_vector_type(8)));

#define HDIM 4096            // 16^3
#define ROWS_PER_BLOCK 4     // 4 waves * 16KB LDS slice = 64KB dynamic LDS
#define THREADS (ROWS_PER_BLOCK * 32)

__global__ __launch_bounds__(THREADS)
void fwht4096_wmma_kernel(const float* __restrict__ x,
                          float* __restrict__ y,
                          int nrows) {
  extern __shared__ float smem[];

  const int lane = threadIdx.x & 31;
  const int wave = threadIdx.x >> 5;
  const int mrow = lane & 15;      // M/N index within a 16-wide half-wave
  const int hi   = lane >> 4;      // half-wave select

  const int row = blockIdx.x * ROWS_PER_BLOCK + wave;
  if (row >= nrows) return;        // wave-uniform (grid divides exactly)

  float* S = smem + wave * HDIM;   // wave-private slice (16 KB)

  // ---- async global -> LDS: 32 x GLOBAL_LOAD_ASYNC_TO_LDS_B128 ---------
  // Flat->LDS mapping truncates to addr[31:0], so the low 32 bits of the
  // generic shared-memory pointer form the VDST LDS-address operand.
  {
    uint32_t lds = (uint32_t)(uintptr_t)S + (uint32_t)lane * 16u;
    uint64_t ga  = (uint64_t)(uintptr_t)(x + (size_t)row * HDIM)
                 + (uint64_t)lane * 16u;
#pragma unroll
    for (int i = 0; i < 32; ++i) {           // 512 B per wave-iteration
      uint32_t l = lds + (uint32_t)i * 512u;
      uint64_t a = ga  + (uint64_t)i * 512u;
      asm volatile("global_load_async_to_lds_b128 %0, %1, off"
                   :: "v"(l), "v"(a) : "memory");
    }
  }

  // ---- B operand: H_16 in registers (and a 1/64-scaled copy) -----------
  // 4x16 B fragment (2 VGPRs): VGPR0 holds K={0,2}, VGPR1 K={1,3}, split
  // across half-waves. H_16[k][n] = (-1)^popcount(k&n). Pinned with empty
  // asm so the compiler keeps them live instead of rematerializing.
  v2f B[4], B3[4];
#pragma unroll
  for (int kk = 0; kk < 4; ++kk) {
    const int k0 = kk * 4 + 2 * hi;
    B[kk].x = (__builtin_popcount(k0 & mrow) & 1) ? -1.0f : 1.0f;
    B[kk].y = (__builtin_popcount((k0 + 1) & mrow) & 1) ? -1.0f : 1.0f;
    B3[kk].x = B[kk].x * 0.015625f;          // fold 1/sqrt(4096) into stage 3
    B3[kk].y = B[kk].y * 0.015625f;
    asm volatile("" : "+v"(B[kk]));
    asm volatile("" : "+v"(B3[kk]));
  }

  // All async copies must have landed in LDS before stage 1 reads.
  asm volatile("s_wait_asynccnt 0" ::: "memory");

  // Row element X[a][b][c], j = a*256 + b*16 + c, a,b,c in [0,16).

  // ---- stage 1: contract c.  tile m = a, M = b, N = c' -----------------
  for (int m = 0; m < 16; ++m) {
    v2f A[4];
    const int abase = m * 256 + mrow * 16 + 2 * hi;      // 8B-aligned
#pragma unroll
    for (int kk = 0; kk < 4; ++kk)
      A[kk] = *reinterpret_cast<const v2f*>(&S[abase + kk * 4]);
    v8f acc = {};
#pragma unroll
    for (int kk = 0; kk < 4; ++kk)
      acc = __builtin_amdgcn_wmma_f32_16x16x4_f32(false, A[kk], false, B[kk],
                                                  (short)0, acc, false, false);
#pragma unroll
    for (int r = 0; r < 8; ++r)
      S[m * 256 + (r + 8 * hi) * 16 + mrow] = acc[r];    // y1[a=m][b=M][c'=N]
  }

  // ---- stage 2: contract b.  tile m = a, M = c', N = b' ----------------
  for (int m = 0; m < 16; ++m) {
    v2f A[4];
#pragma unroll
    for (int kk = 0; kk < 4; ++kk) {
      const int k0 = kk * 4 + 2 * hi;
      A[kk].x = S[m * 256 + k0 * 16 + mrow];
      A[kk].y = S[m * 256 + (k0 + 1) * 16 + mrow];
    }
    v8f acc = {};
#pragma unroll
    for (int kk = 0; kk < 4; ++kk)
      acc = __builtin_amdgcn_wmma_f32_16x16x4_f32(false, A[kk], false, B[kk],
                                                  (short)0, acc, false, false);
    // y2[a=m][b'=N][c'=M]; D row r lands contiguously per lane
    float* p = &S[m * 256 + mrow * 16 + 8 * hi];         // 32B-aligned
    *reinterpret_cast<float4*>(p)     = make_float4(acc[0], acc[1], acc[2], acc[3]);
    *reinterpret_cast<float4*>(p + 4) = make_float4(acc[4], acc[5], acc[6], acc[7]);
  }

  // ---- stage 3: contract a.  tile m = b', M = c', N = a' ---------------
  // Output j' = a'*256 + b'*16 + c' = mrow*256 + m*16 + (r + 8*hi) per lane:
  // contiguous 8 floats -> store straight to global, scale folded into B3.
  {
    float* ybase = y + (size_t)row * HDIM;
    for (int m = 0; m < 16; ++m) {
      v2f A[4];
#pragma unroll
      for (int kk = 0; kk < 4; ++kk) {
        const int k0 = kk * 4 + 2 * hi;
        A[kk].x = S[k0 * 256 + m * 16 + mrow];
        A[kk].y = S[(k0 + 1) * 256 + m * 16 + mrow];
      }
      v8f acc = {};
#pragma unroll
      for (int kk = 0; kk < 4; ++kk)
        acc = __builtin_amdgcn_wmma_f32_16x16x4_f32(false, A[kk], false, B3[kk],
                                                    (short)0, acc, false, false);
      float* p = ybase + mrow * 256 + m * 16 + 8 * hi;   // 32B-aligned
      *reinterpret_cast<float4*>(p)     = make_float4(acc[0], acc[1], acc[2], acc[3]);
      *reinterpret_cast<float4*>(p + 4) = make_float4(acc[4], acc[5], acc[6], acc[7]);
    }
  }
}

extern "C" void kernel_launch(void* const* d_in, const int* in_sizes, int n_in,
                              void* d_out, int out_size, void* d_ws, size_t ws_size,
                              hipStream_t stream) {
  const float* x = (const float*)d_in[0];
  float* y = (float*)d_out;

  const int nrows  = in_sizes[0] / HDIM;                     // 8192
  const int blocks = (nrows + ROWS_PER_BLOCK - 1) / ROWS_PER_BLOCK;
  const size_t shmem = (size_t)ROWS_PER_BLOCK * HDIM * sizeof(float);  // 64 KB

  fwht4096_wmma_kernel<<<blocks, THREADS, shmem, stream>>>(x, y, nrows);
}